// EncoderS5_22093311770900
// MI455X (gfx1250) — compile-verified
//
#include <hip/hip_runtime.h>
#include <hip/hip_bf16.h>
#include <math.h>

// ---------------- problem dims (fixed by reference) ----------------
#define BB   128
#define TT   1024
#define OBS  512
#define HH   64
#define PP   128
#define LL   2
#define CH   16          // timesteps per fused chunk
#define MTOT (BB * TT)   // 131072 GEMM rows

typedef float v2f __attribute__((ext_vector_type(2)));
typedef float v8f __attribute__((ext_vector_type(8)));
typedef int   v4i __attribute__((vector_size(4 * sizeof(int))));

// ---------------- CDNA5 feature probes ----------------
#if defined(__HIP_DEVICE_COMPILE__) && \
    __has_builtin(__builtin_amdgcn_global_load_async_to_lds_b128) && \
    __has_builtin(__builtin_amdgcn_s_wait_asynccnt)
#define USE_ASYNC_LDS 1
#else
#define USE_ASYNC_LDS 0
#endif

// async-copy builtin wants: (addrspace(1) v4i* src, addrspace(3) v4i* dst, imm, imm)
#define AS1V4(p) ((__attribute__((address_space(1))) v4i*)(p))
#define AS3V4(p) ((__attribute__((address_space(3))) v4i*)(p))

// ---------------- WMMA f32 16x16x4 helpers (ISA 7.12.2 layouts) ----------------
__device__ __forceinline__ v8f wmma4(v2f a, v2f b, v8f c) {
    // 8 args: (neg_a, A, neg_b, B, c_mod, C, reuse_a, reuse_b)
    return __builtin_amdgcn_wmma_f32_16x16x4_f32(false, a, false, b, (short)0, c,
                                                 false, false);
}

// A-fragment: A[m][k], m = m0 + (lane&15), k = k0 + 2*(lane>>4) + {0,1}
__device__ __forceinline__ v2f frag_a(const float* s, int m0, int k0, int ld) {
    const int lane = threadIdx.x & 31;
    const int r16 = lane & 15, hf = lane >> 4;
    const float* p = s + (m0 + r16) * ld + k0 + 2 * hf;
    v2f r; r.x = p[0]; r.y = p[1];
    return r;
}

// B-fragment: B[k][n], n = n0 + (lane&15), k = k0 + 2*(lane>>4) + {0,1}
__device__ __forceinline__ v2f frag_b(const float* s, int k0, int n0, int ld) {
    const int lane = threadIdx.x & 31;
    const int r16 = lane & 15, hf = lane >> 4;
    const float* p = s + (k0 + 2 * hf) * ld + n0 + r16;
    v2f r; r.x = p[0]; r.y = p[ld];
    return r;
}

__device__ __forceinline__ v8f vzero8() {
    v8f z = {0.f, 0.f, 0.f, 0.f, 0.f, 0.f, 0.f, 0.f};
    return z;
}

__device__ __forceinline__ float gelu_tanh(float x) {
    const float k0 = 0.7978845608028654f;   // sqrt(2/pi)
    float u = k0 * (x + 0.044715f * x * x * x);
    return 0.5f * x * (1.0f + tanhf(u));
}

// =====================================================================
// Kernel 0: precompute complex discretized weights into workspace.
//   B_bar[p][h] = ((exp(Lam*step)-1)/Lam) * B[p][h]   (stored [h][p] re/im)
//   Ct[p][h]    = C[h][p]  (im part pre-negated; stored [p][h])
//   Lam_bar[p]  = exp(Lam*step)
// grid = (L), block = (P)
// =====================================================================
__global__ __launch_bounds__(PP) void s5_prep_kernel(
    const float* __restrict__ Lre, const float* __restrict__ Lim,
    const float* __restrict__ Bp,  const float* __restrict__ Cp,
    const float* __restrict__ lstep,
    float* __restrict__ wBre, float* __restrict__ wBim,
    float* __restrict__ wCre, float* __restrict__ wCin,
    float* __restrict__ wLr,  float* __restrict__ wLi)
{
    const int l = blockIdx.x, p = threadIdx.x;
    const int lp = l * PP + p;
    const float step = expf(lstep[lp]);
    const float lr = Lre[lp], li = Lim[lp];
    const float er = expf(lr * step);
    const float lbr = er * cosf(li * step);
    const float lbi = er * sinf(li * step);
    const float den = lr * lr + li * li;
    const float nr = lbr - 1.0f, ni = lbi;
    const float cr = (nr * lr + ni * li) / den;   // (Lam_bar-1)/Lam
    const float ci = (ni * lr - nr * li) / den;
    wLr[lp] = lbr;
    wLi[lp] = lbi;
    for (int h = 0; h < HH; ++h) {
        const float br = Bp[(lp * HH + h) * 2 + 0];
        const float bi = Bp[(lp * HH + h) * 2 + 1];
        wBre[l * HH * PP + h * PP + p] = cr * br - ci * bi;
        wBim[l * HH * PP + h * PP + p] = cr * bi + ci * br;
        wCre[l * PP * HH + p * HH + h] =  Cp[((l * HH + h) * PP + p) * 2 + 0];
        wCin[l * PP * HH + p * HH + h] = -Cp[((l * HH + h) * PP + p) * 2 + 1];
    }
}

// =====================================================================
// Kernel 1: x = obs @ dense_w + dense_b   (fp32 WMMA 16x16x4)
// grid = MTOT/128 blocks, 256 threads (8 waves). Wave w owns rows
// [m_base + 16w, +16) x all 64 cols (4 C-tiles). K staged in 64-wide
// LDS slices (obs slice async-copied to LDS on CDNA5).
// =====================================================================
__global__ __launch_bounds__(256) void dense_gemm_kernel(
    const float* __restrict__ obs, const float* __restrict__ w,
    const float* __restrict__ bias, float* __restrict__ x)
{
    __shared__ float sA[128 * 68];   // obs slice, row-padded (+4) for A-frags & b128 align
    __shared__ float sB[64 * 64];    // weight slice [k][n]

    const int tid = threadIdx.x;
    const int lane = tid & 31, wave = tid >> 5;
    const int r16 = lane & 15, hf = lane >> 4;
    const long mbase = (long)blockIdx.x * 128;
    const int m0 = wave * 16;

    v8f acc[4];
#pragma unroll
    for (int nt = 0; nt < 4; ++nt) acc[nt] = vzero8();

    for (int kb = 0; kb < OBS; kb += 64) {
        // prefetch next obs slice into GL2 (global_prefetch_b8)
        if (kb + 64 < OBS)
            __builtin_prefetch(obs + (mbase + (tid >> 1)) * OBS + kb + 64 + (tid & 1) * 32, 0, 1);

        // stage obs slice: 128 rows x 64 K, as 2048 x b128 transfers
#if USE_ASYNC_LDS
#pragma unroll
        for (int it = 0; it < 8; ++it) {
            const int q = tid + it * 256;
            const int r = q >> 4, c4 = (q & 15) * 4;
            __builtin_amdgcn_global_load_async_to_lds_b128(
                AS1V4(obs + (mbase + r) * OBS + kb + c4),
                AS3V4(sA + r * 68 + c4), 0, 0);
        }
#else
#pragma unroll
        for (int it = 0; it < 8; ++it) {
            const int q = tid + it * 256;
            const int r = q >> 4, c4 = (q & 15) * 4;
            *(float4*)(sA + r * 68 + c4) =
                *(const float4*)(obs + (mbase + r) * OBS + kb + c4);
        }
#endif
        // stage weight slice 64x64
#pragma unroll
        for (int it = 0; it < 4; ++it) {
            const int q = tid + it * 256;
            const int r = q >> 4, c4 = (q & 15) * 4;
            *(float4*)(sB + r * 64 + c4) = *(const float4*)(w + (kb + r) * HH + c4);
        }
#if USE_ASYNC_LDS
        __builtin_amdgcn_s_wait_asynccnt(0);
#endif
        __syncthreads();

#pragma unroll
        for (int k = 0; k < 64; k += 4) {
            const v2f a = frag_a(sA, m0, k, 68);
#pragma unroll
            for (int nt = 0; nt < 4; ++nt) {
                const v2f b = frag_b(sB, k, nt * 16, 64);
                acc[nt] = wmma4(a, b, acc[nt]);
            }
        }
        __syncthreads();
    }

#pragma unroll
    for (int nt = 0; nt < 4; ++nt) {
        const int n = nt * 16 + r16;
        const float bn = bias[n];
#pragma unroll
        for (int v = 0; v < 8; ++v) {
            const long row = mbase + m0 + v + 8 * hf;
            x[row * HH + n] = acc[nt][v] + bn;
        }
    }
}

// =====================================================================
// Kernel 2 (per layer, fully fused): layernorm -> Bu WMMA -> reset-scan
// -> C-projection WMMA -> gelu -> GLU WMMA -> sigmoid*y + skip, in-place
// on x. grid = B blocks, 128 threads (4 waves). All weights + all chunk
// intermediates live in LDS (~177 KB of the WGP's 320 KB); Bu/xs never
// touch HBM (saves ~1 GB of round trips vs. materializing).
// =====================================================================
__global__ __launch_bounds__(PP) void s5_layer_kernel(
    float* __restrict__ x,             // (B,T,H), updated in place
    const int* __restrict__ dones,     // (B,T)
    const float* __restrict__ h0,      // (B,P,2) this layer
    const float* __restrict__ wBre,    // (H,P)
    const float* __restrict__ wBim,    // (H,P)
    const float* __restrict__ wCre,    // (P,H)
    const float* __restrict__ wCin,    // (P,H)  (pre-negated imag)
    const float* __restrict__ wLr, const float* __restrict__ wLi,   // (P)
    const float* __restrict__ Dv,
    const float* __restrict__ nsc, const float* __restrict__ nbi,
    const float* __restrict__ gw,      // (H,H)
    const float* __restrict__ gb,
    float* __restrict__ hout)          // (B,P,2) this layer
{
    __shared__ float sBre[HH * PP], sBim[HH * PP];     // Bu weights  [h][p]
    __shared__ float sCre[PP * HH], sCin[PP * HH];     // C weights   [p][h]
    __shared__ float sglu[HH * HH];                    // GLU weights [h][n]
    __shared__ float sxn[CH * 66];                     // layernormed chunk (padded)
    __shared__ float sraw[CH * 66];                    // skip chunk (padded)
    __shared__ float sy[CH * 66];                      // gelu output (padded)
    __shared__ float sstr[CH * 130], ssti[CH * 130];   // Bu -> xs, re/im (padded)
    __shared__ float smu[CH], srs[CH], sdone[CH];

    const int tid = threadIdx.x;
    const int b = blockIdx.x;
    const int lane = tid & 31, wave = tid >> 5;
    const int r16 = lane & 15, hf = lane >> 4;

    // stage all per-layer weights into LDS once
    for (int i = tid; i < HH * PP; i += PP) { sBre[i] = wBre[i]; sBim[i] = wBim[i]; }
    for (int i = tid; i < PP * HH; i += PP) { sCre[i] = wCre[i]; sCin[i] = wCin[i]; }
    for (int i = tid; i < HH * HH; i += PP) sglu[i] = gw[i];

    // per-thread recurrent state: thread tid <-> state index p
    float hre = h0[(b * PP + tid) * 2 + 0];
    float him = h0[(b * PP + tid) * 2 + 1];
    const float lamr = wLr[tid], lami = wLi[tid];
    __syncthreads();

    for (int t0 = 0; t0 < TT; t0 += CH) {
        // ---- 1) load x chunk + resets ----
        const float* xch = x + ((long)b * TT + t0) * HH;
#pragma unroll
        for (int it = 0; it < (CH * HH) / PP; ++it) {
            const int i = tid + it * PP;
            const int t = i >> 6, h = i & 63;
            sraw[t * 66 + h] = xch[i];
        }
        if (tid < CH) sdone[tid] = (float)dones[(long)b * TT + t0 + tid];
        __syncthreads();

        // ---- 2) layernorm ----
        if (tid < CH) {
            float s = 0.f, s2 = 0.f;
            for (int h = 0; h < HH; ++h) {
                const float v = sraw[tid * 66 + h];
                s += v; s2 += v * v;
            }
            const float mu = s * (1.0f / HH);
            const float var = s2 * (1.0f / HH) - mu * mu;
            smu[tid] = mu;
            srs[tid] = rsqrtf(var + 1e-6f);
        }
        __syncthreads();
#pragma unroll
        for (int it = 0; it < (CH * HH) / PP; ++it) {
            const int i = tid + it * PP;
            const int t = i >> 6, h = i & 63;
            sxn[t * 66 + h] = (sraw[t * 66 + h] - smu[t]) * srs[t] * nsc[h] + nbi[h];
        }
        __syncthreads();

        // ---- 3) Bu = xn @ B_bar^T  (re & im), wave owns p-tiles 32w,32w+16 ----
        {
            const int p0 = wave * 32;
            v8f ar0 = vzero8(), ar1 = vzero8(), ai0 = vzero8(), ai1 = vzero8();
#pragma unroll
            for (int k = 0; k < HH; k += 4) {
                const v2f a = frag_a(sxn, 0, k, 66);
                ar0 = wmma4(a, frag_b(sBre, k, p0,      PP), ar0);
                ar1 = wmma4(a, frag_b(sBre, k, p0 + 16, PP), ar1);
                ai0 = wmma4(a, frag_b(sBim, k, p0,      PP), ai0);
                ai1 = wmma4(a, frag_b(sBim, k, p0 + 16, PP), ai1);
            }
#pragma unroll
            for (int v = 0; v < 8; ++v) {
                const int t = v + 8 * hf;
                sstr[t * 130 + p0 + r16]      = ar0[v];
                sstr[t * 130 + p0 + 16 + r16] = ar1[v];
                ssti[t * 130 + p0 + r16]      = ai0[v];
                ssti[t * 130 + p0 + 16 + r16] = ai1[v];
            }
        }
        __syncthreads();

        // ---- 4) sequential reset-scan over the chunk (thread = state p) ----
        {
            const int p = tid;
#pragma unroll
            for (int t = 0; t < CH; ++t) {
                const float br = sstr[t * 130 + p];
                const float bi = ssti[t * 130 + p];
                float nr, ni;
                if (sdone[t] > 0.5f) { nr = br; ni = bi; }
                else {
                    nr = lamr * hre - lami * him + br;
                    ni = lamr * him + lami * hre + bi;
                }
                hre = nr; him = ni;
                sstr[t * 130 + p] = nr;
                ssti[t * 130 + p] = ni;
            }
        }
        __syncthreads();

        // ---- 5) ys = 2*Re(xs @ C^T) + D*xn, gelu; wave owns h-tile 16w ----
        {
            const int n0 = wave * 16;
            v8f acc = vzero8();
#pragma unroll
            for (int k = 0; k < PP; k += 4) {
                acc = wmma4(frag_a(sstr, 0, k, 130), frag_b(sCre, k, n0, HH), acc);
                acc = wmma4(frag_a(ssti, 0, k, 130), frag_b(sCin, k, n0, HH), acc);
            }
            const int h = n0 + r16;
            const float dv = Dv[h];
#pragma unroll
            for (int v = 0; v < 8; ++v) {
                const int t = v + 8 * hf;
                const float yv = 2.0f * acc[v] + dv * sxn[t * 66 + h];
                sy[t * 66 + h] = gelu_tanh(yv);
            }
        }
        __syncthreads();

        // ---- 6) GLU: out = skip + y * sigmoid(y@gw + gb), in-place store ----
        {
            const int n0 = wave * 16;
            v8f acc = vzero8();
#pragma unroll
            for (int k = 0; k < HH; k += 4)
                acc = wmma4(frag_a(sy, 0, k, 66), frag_b(sglu, k, n0, HH), acc);
            const int n = n0 + r16;
            const float bias = gb[n];
            float* xo = x + ((long)b * TT + t0) * HH;
#pragma unroll
            for (int v = 0; v < 8; ++v) {
                const int t = v + 8 * hf;
                const float g = acc[v] + bias;
                const float sig = 1.0f / (1.0f + expf(-g));
                xo[t * HH + n] = sraw[t * 66 + n] + sy[t * 66 + n] * sig;
            }
        }
        __syncthreads();
    }

    hout[(b * PP + tid) * 2 + 0] = hre;
    hout[(b * PP + tid) * 2 + 1] = him;
}

// =====================================================================
// Host launcher
// =====================================================================
extern "C" void kernel_launch(void* const* d_in, const int* in_sizes, int n_in,
                              void* d_out, int out_size, void* d_ws, size_t ws_size,
                              hipStream_t stream) {
    const float* obs     = (const float*)d_in[0];
    const int*   dones   = (const int*)  d_in[1];
    const float* hidden  = (const float*)d_in[2];
    const float* dense_w = (const float*)d_in[3];
    const float* dense_b = (const float*)d_in[4];
    const float* Lre     = (const float*)d_in[5];
    const float* Lim     = (const float*)d_in[6];
    const float* Bp      = (const float*)d_in[7];
    const float* Cp      = (const float*)d_in[8];
    const float* Dv      = (const float*)d_in[9];
    const float* lstep   = (const float*)d_in[10];
    const float* nsc     = (const float*)d_in[11];
    const float* nbi     = (const float*)d_in[12];
    const float* gw      = (const float*)d_in[13];
    const float* gb      = (const float*)d_in[14];

    float* out   = (float*)d_out;
    float* h_out = out;                               // L*B*P*2 = 65536 floats
    float* x     = out + (size_t)LL * BB * PP * 2;    // B*T*H, working buffer + result

    // workspace layout (264 KB total)
    float* ws    = (float*)d_ws;
    float* wsBre = ws;                                // L * H*P
    float* wsBim = wsBre + (size_t)LL * HH * PP;
    float* wsCre = wsBim + (size_t)LL * HH * PP;
    float* wsCin = wsCre + (size_t)LL * PP * HH;
    float* wsLr  = wsCin + (size_t)LL * PP * HH;      // L * P
    float* wsLi  = wsLr + (size_t)LL * PP;

    s5_prep_kernel<<<dim3(LL), dim3(PP), 0, stream>>>(
        Lre, Lim, Bp, Cp, lstep, wsBre, wsBim, wsCre, wsCin, wsLr, wsLi);

    dense_gemm_kernel<<<dim3(MTOT / 128), dim3(256), 0, stream>>>(
        obs, dense_w, dense_b, x);

    for (int l = 0; l < LL; ++l) {
        s5_layer_kernel<<<dim3(BB), dim3(PP), 0, stream>>>(
            x, dones,
            hidden + (size_t)l * BB * PP * 2,
            wsBre + (size_t)l * HH * PP, wsBim + (size_t)l * HH * PP,
            wsCre + (size_t)l * PP * HH, wsCin + (size_t)l * PP * HH,
            wsLr + (size_t)l * PP, wsLi + (size_t)l * PP,
            Dv + (size_t)l * HH, nsc + (size_t)l * HH, nbi + (size_t)l * HH,
            gw + (size_t)l * HH * HH, gb + (size_t)l * HH,
            h_out + (size_t)l * BB * PP * 2);
    }
}